// CAModel2_27049704030506
// MI455X (gfx1250) — compile-verified
//
#include <hip/hip_runtime.h>

// ---------------------------------------------------------------------------
// Neural-CA step for MI455X (gfx1250), wave32, fp32 WMMA (16x16x4).
//
// Per workgroup (256 threads = 8 waves), block = (b, ww):
//   owns NHWC output region  out[b, h2=0..255, w2=4*ww..4*ww+3, c2=0..15]
//   needs conv of source columns  w in {ww, ww+64, ww+128, ww+192} (+/-1 halo)
//
// Phases:
//   1. stage 12 x-columns (h halo, stride-20 pad) + conv weights in LDS
//   2. conv as implicit GEMM via V_WMMA_F32_16X16X4_F32:
//      D[m=hc, n=h16], K=(a,d,c)=144 -> 36 chained WMMAs per tile.
//      All 36 weight A-fragments preloaded to VGPRs once (no masked loads
//      in the loop); B loads use one base reg + DS immediate offsets and
//      are issued 4-ahead so dscnt waits stagger instead of hitting 0.
//   3. MLP: fc0 = single WMMA (K=4 exact, bias in C) + ReLU;
//      fc1 = 3 accumulating WMMAs (K=12); D-layout -> A/B-layout through
//      ping-ponged per-wave LDS scratch so consecutive tiles overlap.
//   4. out = x + delta (delta LDS aliases x-column region), coalesced float4.
// ---------------------------------------------------------------------------

typedef __attribute__((ext_vector_type(2))) float v2f;
typedef __attribute__((ext_vector_type(8))) float v8f;

#define BATCH  32
#define HEIGHT 256
#define WIDTH  256
#define CHAN   16
#define HCN    4
#define HS     12

// LDS layout (floats)
#define XROW_STRIDE 20                      // 16 channels padded to 20 (conflict-free, 16B aligned)
#define XROWS       258                     // h in [-1, 256]
#define XCOL_SIZE   (XROWS * XROW_STRIDE)   // 5160
#define NCOLS       12                      // 4 source cols x 3 (w halo)
#define OFF_XCOL    0
#define OFF_CONVW   (NCOLS * XCOL_SIZE)     // 61920
#define OFF_CONVOUT (OFF_CONVW + 576)       // 62496: [4 cols][4 hc][256 h]
#define OFF_US      (OFF_CONVOUT + 4096)    // 66592: 8 waves x 2 x 256 (ping-pong)
#define SMEM_FLOATS (OFF_US + 8 * 2 * 256)  // 70688 floats = 282752 B (<= 320KB/WGP)
#define DELTA_STRIDE 260
#define OFF_DELTA   0                       // aliases x-column region

__launch_bounds__(256)
__global__ void ca_step_kernel(const float* __restrict__ xin,
                               const float* __restrict__ convw,
                               const float* __restrict__ fc0w,
                               const float* __restrict__ fc0b,
                               const float* __restrict__ fc1w,
                               float* __restrict__ xout)
{
    extern __shared__ float smem[];
    float* xcol  = smem + OFF_XCOL;
    float* cw    = smem + OFF_CONVW;
    float* cout  = smem + OFF_CONVOUT;
    float* us    = smem + OFF_US;
    float* delta = smem + OFF_DELTA;

    const int tid  = threadIdx.x;
    const int lane = tid & 31;
    const int wave = tid >> 5;
    const int m    = lane & 15;   // striped index (M for A, N for B/C/D)
    const int hi   = lane >> 4;   // K-half selector

    const int blk = blockIdx.x;
    const int b   = blk >> 6;
    const int ww  = blk & 63;

    const float* xres = xin + ((b * HEIGHT) * WIDTH + ww * 4) * CHAN;

    // Early prefetch of the phase-4 residual region (64KB) into cache while
    // the LDS/WMMA phases run.  4 lines per thread.
    for (int i = tid; i < 1024; i += 256) {
        int h2 = i >> 2;
        int q  = i & 3;
        __builtin_prefetch(xres + (h2 * WIDTH + q) * CHAN, 0, 1);
    }

    // ---- per-thread MLP weight preload (global -> regs, tiny & cached) ----
    v2f afc0;                                   // fc0 A[m=s, k=j2]
    {
        int s = m;
        int base = (s < HS ? s : 0) * 4 + 2 * hi;
        float a0 = fc0w[base + 0];
        float a1 = fc0w[base + 1];
        afc0 = (s < HS) ? (v2f){a0, a1} : (v2f){0.f, 0.f};
    }
    v8f cbias;                                  // fc0 C[s, row] = fc0_b[s]
#pragma unroll
    for (int v = 0; v < 8; ++v) {
        int s = v + 8 * hi;
        float bv = fc0b[s < HS ? s : 0];
        cbias[v] = (s < HS) ? bv : 0.f;
    }
    v2f afc1[3];                                // fc1 A[m=c_out, k=s]
#pragma unroll
    for (int t3 = 0; t3 < 3; ++t3) {
        int base = m * HS + 4 * t3 + 2 * hi;
        afc1[t3].x = fc1w[base + 0];
        afc1[t3].y = fc1w[base + 1];
    }

    // ---- phase 1: stage conv weights + 12 x-columns (with halos) in LDS ----
    for (int i = tid; i < 576; i += 256) cw[i] = convw[i];

    for (int i = tid; i < NCOLS * XROWS * 4; i += 256) {
        int colIdx = i / (XROWS * 4);
        int rem    = i - colIdx * (XROWS * 4);
        int hh     = rem >> 2;
        int c4     = rem & 3;
        int t      = colIdx / 3;
        int aoff   = colIdx - t * 3 - 1;              // -1, 0, +1
        int wcol   = ww + 64 * t + aoff;
        int h      = hh - 1;
        float4 val = make_float4(0.f, 0.f, 0.f, 0.f);
        if ((unsigned)h < 256u && (unsigned)wcol < 256u) {
            val = *(const float4*)(xin + ((b * HEIGHT + h) * WIDTH + wcol) * CHAN + c4 * 4);
        }
        *(float4*)(xcol + colIdx * XCOL_SIZE + hh * XROW_STRIDE + c4 * 4) = val;
    }
    __syncthreads();

    // ---- phase 2: conv as implicit GEMM, D[m=hc, n=h], K=(a,d,c)=144 ----

    // Preload ALL 36 weight A-fragments to VGPRs: unconditional clamped
    // loads + one cndmask each.  No EXEC-masked loads in the hot loop.
    v2f aw[36];
    {
        const float* pw = cw + (m & 3) * 144 + (2 * hi) * 9;
#pragma unroll
        for (int a = 0; a < 3; ++a)
#pragma unroll
            for (int d = 0; d < 3; ++d)
#pragma unroll
                for (int cblk = 0; cblk < 4; ++cblk) {
                    float w0 = pw[cblk * 36 + 0 + a * 3 + d];
                    float w1 = pw[cblk * 36 + 9 + a * 3 + d];
                    aw[(a * 3 + d) * 4 + cblk] =
                        (m < HCN) ? (v2f){w0, w1} : (v2f){0.f, 0.f};
                }
    }

    for (int uu = 0; uu < 8; ++uu) {
        int u  = wave * 8 + uu;      // 64 units: 4 cols x 16 h-tiles
        int t  = u >> 4;
        int h0 = (u & 15) << 4;
        v8f acc = {};
        // one base address; every B load is base + compile-time DS offset
        const float* pb = xcol + 3 * t * XCOL_SIZE + (h0 + m) * XROW_STRIDE + 2 * hi;
#pragma unroll
        for (int a = 0; a < 3; ++a) {
#pragma unroll
            for (int d = 0; d < 3; ++d) {
                const float* pg = pb + a * XCOL_SIZE + d * XROW_STRIDE;
                v2f b0 = *(const v2f*)(pg + 0);
                v2f b1 = *(const v2f*)(pg + 4);
                v2f b2 = *(const v2f*)(pg + 8);
                v2f b3 = *(const v2f*)(pg + 12);
                int s = (a * 3 + d) * 4;
                acc = __builtin_amdgcn_wmma_f32_16x16x4_f32(
                    false, aw[s + 0], false, b0, (short)0, acc, false, false);
                acc = __builtin_amdgcn_wmma_f32_16x16x4_f32(
                    false, aw[s + 1], false, b1, (short)0, acc, false, false);
                acc = __builtin_amdgcn_wmma_f32_16x16x4_f32(
                    false, aw[s + 2], false, b2, (short)0, acc, false, false);
                acc = __builtin_amdgcn_wmma_f32_16x16x4_f32(
                    false, aw[s + 3], false, b3, (short)0, acc, false, false);
            }
        }
        // ReLU + store rows hc=0..3 (vgprs 0..3, lanes 0..15), conflict-free
        if (lane < 16) {
            float* cdst = cout + (t * 4) * 256 + h0 + lane;
            cdst[0]   = fmaxf(acc[0], 0.f);
            cdst[256] = fmaxf(acc[1], 0.f);
            cdst[512] = fmaxf(acc[2], 0.f);
            cdst[768] = fmaxf(acc[3], 0.f);
        }
    }
    __syncthreads();

    // ---- phase 3: MLP, 64 row-tiles (hc, t, q); delta aliases x region ----
#pragma unroll 2
    for (int uu = 0; uu < 8; ++uu) {
        int tt = wave * 8 + uu;
        int hc = tt >> 4;
        int t  = (tt >> 2) & 3;
        int q  = tt & 3;

        // fc0: D'[s,row] = fc0_w x V^T + bias ; V[row][j2]=relu(conv[h=(q*16+row)*4+j2])
        v2f Bv = *(const v2f*)(cout + (t * 4 + hc) * 256 + (q * 16 + m) * 4 + 2 * hi);
        v8f u0 = __builtin_amdgcn_wmma_f32_16x16x4_f32(
            false, afc0, false, Bv, (short)0, cbias, false, false);
#pragma unroll
        for (int v = 0; v < 8; ++v) u0[v] = fmaxf(u0[v], 0.f);

        // D-layout -> A/B-layout via ping-ponged per-wave LDS scratch us[row*16+s]
        float* usw = us + wave * 512 + (uu & 1) * 256;
        *(float4*)(usw + m * 16 + 8 * hi + 0) = make_float4(u0[0], u0[1], u0[2], u0[3]);
        *(float4*)(usw + m * 16 + 8 * hi + 4) = make_float4(u0[4], u0[5], u0[6], u0[7]);

        // fc1: D2[j,row] = fc1_w x U^T (K=12 -> 3 accumulating WMMAs)
        v8f acc2 = {};
#pragma unroll
        for (int t3 = 0; t3 < 3; ++t3) {
            v2f B2 = *(const v2f*)(usw + m * 16 + 4 * t3 + 2 * hi);
            acc2 = __builtin_amdgcn_wmma_f32_16x16x4_f32(
                false, afc1[t3], false, B2, (short)0, acc2, false, false);
        }

        // scatter: h2 = row*16 + j ; column (c2=4hc+t, w2=4ww+q)
        int cq = (4 * hc + t) * 4 + q;
        float* dbase = delta + cq * DELTA_STRIDE + m * 16 + 8 * hi;
        *(v2f*)(dbase + 0) = (v2f){acc2[0], acc2[1]};
        *(v2f*)(dbase + 2) = (v2f){acc2[2], acc2[3]};
        *(v2f*)(dbase + 4) = (v2f){acc2[4], acc2[5]};
        *(v2f*)(dbase + 6) = (v2f){acc2[6], acc2[7]};
    }
    __syncthreads();

    // ---- phase 4: out = x + delta, coalesced NHWC float4 ----
    float* outp = xout + ((b * HEIGHT) * WIDTH + ww * 4) * CHAN;
    for (int i = tid; i < 4096; i += 256) {
        int h2  = i >> 4;
        int q   = (i >> 2) & 3;
        int c2b = (i & 3) << 2;
        int goff = (h2 * WIDTH + q) * CHAN + c2b;
        float4 xv = *(const float4*)(xres + goff);
        float4 dv;
        dv.x = delta[((c2b + 0) * 4 + q) * DELTA_STRIDE + h2];
        dv.y = delta[((c2b + 1) * 4 + q) * DELTA_STRIDE + h2];
        dv.z = delta[((c2b + 2) * 4 + q) * DELTA_STRIDE + h2];
        dv.w = delta[((c2b + 3) * 4 + q) * DELTA_STRIDE + h2];
        *(float4*)(outp + goff) =
            make_float4(xv.x + dv.x, xv.y + dv.y, xv.z + dv.z, xv.w + dv.w);
    }
}

extern "C" void kernel_launch(void* const* d_in, const int* in_sizes, int n_in,
                              void* d_out, int out_size, void* d_ws, size_t ws_size,
                              hipStream_t stream) {
    (void)in_sizes; (void)n_in; (void)out_size; (void)ws_size;
    const float* x   = (const float*)d_in[0];
    const float* cwv = (const float*)d_in[1];
    const float* f0w = (const float*)d_in[2];
    const float* f0b = (const float*)d_in[3];
    const float* f1w = (const float*)d_in[4];
    // d_in[5] = steps (== 2, fixed by the reference)

    float* tmp = (float*)d_ws;    // 128 MB ping buffer for the inter-step sync
    float* out = (float*)d_out;

    dim3 grid(BATCH * 64);        // (b, ww): 2048 workgroups
    dim3 block(256);              // 8 waves (wave32)
    size_t shmem = SMEM_FLOATS * sizeof(float);   // 282752 B dynamic LDS

    ca_step_kernel<<<grid, block, shmem, stream>>>(x,   cwv, f0w, f0b, f1w, tmp);
    ca_step_kernel<<<grid, block, shmem, stream>>>(tmp, cwv, f0w, f0b, f1w, out);
}